// JeffressFilter_62715112456224
// MI455X (gfx1250) — compile-verified
//
#include <hip/hip_runtime.h>

// ---------------------------------------------------------------------------
// JeffressFilter for MI455X (gfx1250, wave32).
//
// Stage 1 (lif_scan_wmma): y = M @ x where M[t,s] = decay^(t-s), s<=t.
//   decay = exp(-1/2); decay^k < f32 denorm-min for k > ~207, so M is a
//   banded (bandwidth 256) lower-triangular matrix. We tile time into 64
//   tiles of 16 and compute each output tile as a sum of <=16 16x16 blocks
//   of M against the 32 channels using V_WMMA_F32_16X16X4_F32. Matrix
//   entries are synthesized in-register via v_exp_f32 (no memory for M).
//   One wave per output tile -> no serial recurrence at all.
//
// Stage 2 (jeffress_scatter): out[t,nc,d] = w[d] * ( y[t-d0, nc, 0]*(t>=d0)
//                                               +  y[t-d1, nc, 1]*(t>=d1) ).
//   32 MiB of output @ 23.3 TB/s ~ 1.4us floor; y (128 KB) is L2-resident,
//   stores are fully coalesced float4 along the d axis.
// ---------------------------------------------------------------------------

typedef __attribute__((ext_vector_type(2))) float v2f;
typedef __attribute__((ext_vector_type(8))) float v8f;

#define T_LEN   1024
#define NC      16                         // N*C
#define CH      32                         // N*C*2 interleaved channels
#define D_OUT   512
#define DECAY16 0.00033546262790251185f    // decay^16 = exp(-8)
#define L2DECAY (-0.72134752044448170f)    // log2(exp(-0.5))

// ---- Stage 1: banded lower-triangular matmul scan via f32 WMMA ------------
__global__ __launch_bounds__(128) void lif_scan_wmma(const float* __restrict__ x,
                                                     float* __restrict__ y) {
  const int wave = (blockIdx.x * blockDim.x + threadIdx.x) >> 5;  // 0..63
  const int lane = threadIdx.x & 31;
  const int m    = lane & 15;   // A-matrix row (M dim) held by this lane
  const int hi   = lane >> 4;   // lanes 16-31 hold K = 2,3 of each K-group
  const int tt   = wave;        // output time tile (16 rows)

  // Base A fragments for the dt=0 diagonal block: A[m,k] = decay^(m-k).
  // For dt>0 blocks we just rescale by decay^(16*dt) (all exponents > 0).
  v2f baseA[4];
#pragma unroll
  for (int kk = 0; kk < 4; ++kk) {
    const int k0 = 4 * kk + 2 * hi;
    baseA[kk].x = __builtin_exp2f((float)(m - k0) * L2DECAY);
    baseA[kk].y = __builtin_exp2f((float)(m - k0 - 1) * L2DECAY);
  }

  v8f acc0 = {};   // channels 0..15
  v8f acc1 = {};   // channels 16..31
  float scale = 1.0f;                       // decay^(16*dt)
  const int ndt = (tt < 15) ? (tt + 1) : 16;  // banded: 16 K-tiles max
  for (int dt = 0; dt < ndt; ++dt) {        // uniform per-wave bound
    const int st = tt - dt;                 // source time tile
    const float* xs = x + st * (16 * CH);
#pragma unroll
    for (int kk = 0; kk < 4; ++kk) {
      const int k0 = 4 * kk + 2 * hi;
      v2f a;
      if (dt == 0) {  // triangular mask on the diagonal block
        a.x = (m >= k0)     ? baseA[kk].x : 0.0f;
        a.y = (m >= k0 + 1) ? baseA[kk].y : 0.0f;
      } else {
        a.x = scale * baseA[kk].x;
        a.y = scale * baseA[kk].y;
      }
      // B fragment (4x16): VGPR0 holds K rows {2*hi}, VGPR1 K rows {2*hi+1},
      // lanes 0..15 hold N (=channel) 0..15 of the selected half.
      const float* bp = xs + k0 * CH + m;
      v2f b0; b0.x = bp[0];  b0.y = bp[CH];
      acc0 = __builtin_amdgcn_wmma_f32_16x16x4_f32(
          false, a, false, b0, (short)0, acc0, false, false);
      v2f b1; b1.x = bp[16]; b1.y = bp[CH + 16];
      acc1 = __builtin_amdgcn_wmma_f32_16x16x4_f32(
          false, a, false, b1, (short)0, acc1, false, false);
    }
    scale *= DECAY16;
  }

  // C/D layout: VGPR v, lanes 0-15 -> row v, lanes 16-31 -> row v+8.
  float* yp = y + (tt * 16 + 8 * hi) * CH + m;
#pragma unroll
  for (int v = 0; v < 8; ++v) {
    yp[v * CH]      = acc0[v];
    yp[v * CH + 16] = acc1[v];
  }
}

// ---- Stage 2: store-bandwidth-bound delayed gather + scale ----------------
__global__ __launch_bounds__(256) void jeffress_scatter(const float* __restrict__ y,
                                                        const float* __restrict__ w,
                                                        const int* __restrict__ delay,
                                                        float* __restrict__ out) {
  const int gid = blockIdx.x * blockDim.x + threadIdx.x;  // one float4 of out
  const int d4  = gid & (D_OUT / 4 - 1);                  // 0..127
  const int tn  = gid >> 7;                               // t*16 + nc
  const int nc  = tn & (NC - 1);
  const int t   = tn >> 4;

  const float* yb = y + nc * 2;   // y[t][nc][i] at yb[t*CH + i]
  float4 r;
  float* rp = reinterpret_cast<float*>(&r);
#pragma unroll
  for (int j = 0; j < 4; ++j) {
    const int d  = d4 * 4 + j;
    const int d0 = delay[2 * d];
    const int d1 = delay[2 * d + 1];
    float s = 0.0f;
    const int t0 = t - d0;
    if (t0 >= 0) s += yb[t0 * CH];
    const int t1 = t - d1;
    if (t1 >= 0) s += yb[t1 * CH + 1];
    rp[j] = s * w[d];
  }
  reinterpret_cast<float4*>(out)[gid] = r;   // coalesced 16B store
}

extern "C" void kernel_launch(void* const* d_in, const int* in_sizes, int n_in,
                              void* d_out, int out_size, void* d_ws, size_t ws_size,
                              hipStream_t stream) {
  const float* x     = (const float*)d_in[0];  // (1024, 4, 4, 2) f32
  const float* wgt   = (const float*)d_in[1];  // (512,) f32
  const int*   delay = (const int*)d_in[2];    // (512, 2) i32
  float* out = (float*)d_out;                  // (1024, 4, 4, 512) f32
  float* y   = (float*)d_ws;                   // scratch: 1024*32 f32 = 128 KB

  // 64 waves, one 16-row time tile each (4 waves/block).
  lif_scan_wmma<<<16, 128, 0, stream>>>(x, y);

  // 2,097,152 float4 outputs -> 8192 blocks x 256 threads.
  jeffress_scatter<<<(T_LEN * NC * (D_OUT / 4)) / 256, 256, 0, stream>>>(y, wgt, delay, out);
}